// NaiveBias_30451318128836
// MI455X (gfx1250) — compile-verified
//
#include <hip/hip_runtime.h>

typedef float v2f  __attribute__((ext_vector_type(2)));
typedef float v8f  __attribute__((ext_vector_type(8)));
typedef float f32x4 __attribute__((ext_vector_type(4)));

#define SEQ 2048
#define NH  16
#define DH  64
#define KT  32          // K-tile (rows of v staged in LDS per step)
#define VST 72          // LDS dword stride: 2*72 mod 64 == 16 -> half-waves hit disjoint banks
#define JT  256         // output rows per workgroup (8 waves * 32 rows)
#define WC  2048        // mirror center of w_ext: w_ext[WC + r] = w[|r|]

// ---- CDNA5 async global->LDS copy (ASYNCcnt-tracked, no VGPR round-trip) ----
__device__ __forceinline__ void async_b128(unsigned lds_off, unsigned long long gaddr) {
    asm volatile("global_load_async_to_lds_b128 %0, %1, off"
                 :: "v"(lds_off), "v"(gaddr) : "memory");
}
__device__ __forceinline__ void wait_async0() {
    asm volatile("s_wait_asynccnt 0x0" ::: "memory");
}

__global__ __launch_bounds__(256)
void naivebias_toeplitz_wmma(const float* __restrict__ v,
                             const float* __restrict__ w,
                             float* __restrict__ out)
{
    __shared__ float w_ext[2 * SEQ];            // mirrored w row: w_ext[WC+r]=w[|r|] (16 KB)
    __shared__ float v_lds[2][KT * VST];        // double-buffered v K-tile (2 x 9 KB)

    const int jblocks = SEQ / JT;               // 8
    const int blk = blockIdx.x;
    const int jb  = blk % jblocks;
    const int h   = (blk / jblocks) % NH;
    const int b   = blk / (jblocks * NH);

    const int tid  = threadIdx.x;
    const int wave = tid >> 5;
    const int lane = tid & 31;

    // ---- stage mirrored w row into LDS ----
    {
        const float* wrow = w + h * SEQ;
        // upper half [WC, WC+2047] = w[0..2047], vectorized (16B-aligned dst)
        const f32x4* wsrc = (const f32x4*)wrow;
        f32x4* wdst = (f32x4*)(w_ext + WC);
        wdst[tid]       = wsrc[tid];
        wdst[tid + 256] = wsrc[tid + 256];
        // mirror half [1, WC]: w_ext[e] = w[WC - e]  (8 scalars/thread, once per block)
        #pragma unroll
        for (int i = 0; i < 8; ++i) {
            const int e = tid * 8 + i + 1;      // 1..2048
            w_ext[e] = wrow[WC - e];
        }
    }

    const int j0   = jb * JT + wave * 32;       // this wave's first output row
    const int mrow = lane & 15;                 // M row / N col within a 16x16 tile
    const int kp   = (lane >> 4) * 2;           // K sub-offset for this half-wave (0 or 2)

    // per-lane affine A-fragment bases: index = abase + l0 + kk  (kk folds to imm offsets)
    const int abase0 = WC + kp - j0 - mrow;     // M-tile 0
    const int abase1 = abase0 - 16;             // M-tile 1

    v8f acc[2][4];
    #pragma unroll
    for (int mt = 0; mt < 2; ++mt)
        #pragma unroll
        for (int nt = 0; nt < 4; ++nt)
            acc[mt][nt] = 0.0f;

    // per-thread staging coordinates (2048 floats / 256 thr = 8 floats = 2 x b128 each)
    const int skk = tid >> 3;                   // 0..31 (row within tile)
    const int sd  = (tid & 7) * 8;              // 0..56 (col base)
    const float* vbase = v + (((size_t)b * SEQ) * NH + h) * DH;   // + l*NH*DH + d

    const unsigned lds_dst0 = (unsigned)(uintptr_t)&v_lds[0][skk * VST + sd];
    const unsigned lds_dst1 = (unsigned)(uintptr_t)&v_lds[1][skk * VST + sd];

    // prologue: kick off tile 0
    {
        unsigned long long g = (unsigned long long)(uintptr_t)
            (vbase + (size_t)skk * (NH * DH) + sd);
        async_b128(lds_dst0,      g);
        async_b128(lds_dst0 + 16, g + 16);
    }

    const int NT = SEQ / KT;                    // 64 K-tiles
    for (int t = 0; t < NT; ++t) {
        const int cur = t & 1;
        wait_async0();                          // my loads for buf[cur] have landed
        __syncthreads();                        // everyone's landed; prev buffer fully consumed

        if (t + 1 < NT) {                       // overlap next tile's copy with compute
            unsigned long long g = (unsigned long long)(uintptr_t)
                (vbase + (size_t)((t + 1) * KT + skk) * (NH * DH) + sd);
            const unsigned dst = cur ? lds_dst0 : lds_dst1;
            async_b128(dst,      g);
            async_b128(dst + 16, g + 16);
        }

        const float* vb  = v_lds[cur];
        const int    l0  = t * KT;
        const float* wa0 = w_ext + abase0 + l0; // per-tile A bases (affine update)
        const float* wa1 = w_ext + abase1 + l0;

        #pragma unroll
        for (int kk = 0; kk < KT; kk += 4) {
            // A fragments: adjacent mirrored-Toeplitz values -> ds_load_2addr candidates
            v2f afrag[2];
            afrag[0][0] = wa0[kk];
            afrag[0][1] = wa0[kk + 1];
            afrag[1][0] = wa1[kk];
            afrag[1][1] = wa1[kk + 1];

            // B fragments: v tile columns from padded LDS
            v2f bfrag[4];
            #pragma unroll
            for (int nt = 0; nt < 4; ++nt) {
                const float* bp = vb + (kk + kp) * VST + nt * 16 + mrow;
                bfrag[nt][0] = bp[0];
                bfrag[nt][1] = bp[VST];
            }

            // 8 fp32 WMMAs per K=4 step (16K FLOP)
            #pragma unroll
            for (int mt = 0; mt < 2; ++mt)
                #pragma unroll
                for (int nt = 0; nt < 4; ++nt)
                    acc[mt][nt] = __builtin_amdgcn_wmma_f32_16x16x4_f32(
                        false, mt ? afrag[1] : afrag[0],
                        false, bfrag[nt],
                        (short)0, acc[mt][nt], false, false);
        }
    }

    // ---- write back: C/D layout = VGPR r -> rows r (lanes 0-15) / r+8 (lanes 16-31) ----
    const size_t obase = (((size_t)b * SEQ) * NH + h) * DH;
    const int rext = (lane < 16) ? 0 : 8;
    const int dcol = lane & 15;
    #pragma unroll
    for (int mt = 0; mt < 2; ++mt) {
        #pragma unroll
        for (int r = 0; r < 8; ++r) {
            const int j = j0 + mt * 16 + r + rext;
            float* orow = out + obase + (size_t)j * (NH * DH) + dcol;
            #pragma unroll
            for (int nt = 0; nt < 4; ++nt)
                orow[nt * 16] = acc[mt][nt][r];
        }
    }
}

extern "C" void kernel_launch(void* const* d_in, const int* in_sizes, int n_in,
                              void* d_out, int out_size, void* d_ws, size_t ws_size,
                              hipStream_t stream) {
    const float* v = (const float*)d_in[0];   // (B, S, H, D) fp32
    const float* w = (const float*)d_in[1];   // (1, H, S)    fp32
    float* out = (float*)d_out;               // (B, S, H, D) fp32

    const int B = in_sizes[0] / (SEQ * NH * DH);       // 2
    const int blocks = B * NH * (SEQ / JT);            // 256
    naivebias_toeplitz_wmma<<<blocks, 256, 0, stream>>>(v, w, out);
}